// TopK_10548439679699
// MI455X (gfx1250) — compile-verified
//
#include <hip/hip_runtime.h>
#include <cstdint>

// Row-wise top-K(=64) + ReLU-on-kept + scatter-into-zeros.
// One workgroup (1024 threads = 32 wave32) per row of 32768 fp32.
//
// Pipeline per row:
//   1. DMA the whole 128KB row into LDS with global_load_async_to_lds_b128
//      (ASYNCcnt path; in-order completion lets us consume chunk j after
//      s_wait_asynccnt(7-j), overlapping DMA with pass-1 histogramming).
//   2. Each thread reads its 32 elements from LDS ONCE, converts them to
//      order-preserving 32-bit keys held in registers.
//   3. 3-pass radix select (11/11/10-bit digits, 2048-bin LDS histogram with
//      ds_add atomics) finds the exact 64th-largest key T per row.
//   4. Write pass reconstructs floats from register keys and emits
//      relu(x) where key>T (+ tie budget at ==T), 0 elsewhere, with
//      nontemporal b128 stores (output is a write-once stream).
//
// HBM traffic is the roofline minimum: 1 read + 1 write per element
// (~1GB @ 23.3 TB/s ~ 45us).

#define NCOLS 32768
#define BLOCK 1024
#define VEC_PER_THREAD 8   // 8 x float4 = 32 elements per thread
#define KSEL 64u

typedef int   v4i __attribute__((ext_vector_type(4)));
typedef float v4f __attribute__((ext_vector_type(4)));
typedef __attribute__((address_space(1))) v4i* gv4i_p;   // global 16B vector ptr
typedef __attribute__((address_space(3))) v4i* lv4i_p;   // LDS 16B vector ptr

#if __has_builtin(__builtin_amdgcn_global_load_async_to_lds_b128)
  #define HAVE_ASYNC_LDS 1
#else
  #define HAVE_ASYNC_LDS 0
#endif

#if HAVE_ASYNC_LDS
  #if __has_builtin(__builtin_amdgcn_s_wait_asynccnt)
    #define WAIT_ASYNC(N) __builtin_amdgcn_s_wait_asynccnt(N)
  #else
    #define WAIT_ASYNC(N) asm volatile("s_wait_asynccnt %0" :: "i"(N) : "memory")
  #endif
#else
  #define WAIT_ASYNC(N) do {} while (0)
#endif

__device__ __forceinline__ unsigned f2key(float f) {
  unsigned u = __float_as_uint(f);
  // order-preserving map: larger float -> larger unsigned
  return (u & 0x80000000u) ? ~u : (u | 0x80000000u);
}

__device__ __forceinline__ float key2f(unsigned k) {
  const unsigned u = (k & 0x80000000u) ? (k & 0x7fffffffu) : ~k;
  return __uint_as_float(u);
}

// Find bin (scanning from the top) where the suffix count first reaches `want`.
// hist has `nb` bins (1024 or 2048). Accumulates the strictly-above count into
// *s_above, writes the selected bin to *s_bin.
__device__ __forceinline__ void select_bin(unsigned nb, unsigned want,
                                           unsigned* hist, unsigned* chunk,
                                           unsigned* s_bin, unsigned* s_above,
                                           int tid) {
  const unsigned per = nb >> 6;  // bins per chunk (64 chunks)
  if (tid < 64) {
    unsigned s = 0;
    for (unsigned i = 0; i < per; ++i) s += hist[(unsigned)tid * per + i];
    chunk[tid] = s;
  }
  __syncthreads();
  if (tid == 0) {
    unsigned above = 0;
    int c = 63;
    for (; c > 0; --c) {
      unsigned s = chunk[c];
      if (above + s >= want) break;
      above += s;
    }
    const unsigned base = (unsigned)c * per;
    int b = (int)per - 1;
    for (; b > 0; --b) {
      unsigned s = hist[base + (unsigned)b];
      if (above + s >= want) break;
      above += s;
    }
    *s_bin = base + (unsigned)b;
    *s_above += above;  // strictly-above-selected-bin, accumulated across passes
  }
  __syncthreads();
}

__global__ __launch_bounds__(BLOCK)
void topk_relu_scatter(const float* __restrict__ x, float* __restrict__ out) {
  __shared__ __align__(16) float tile[NCOLS];   // 128 KB staging buffer
  __shared__ unsigned hist[2048];               // 8 KB
  __shared__ unsigned chunk[64];
  __shared__ unsigned s_bin, s_above, s_claim;

  const int tid = (int)threadIdx.x;
  const size_t rowBase = (size_t)blockIdx.x * (size_t)NCOLS;
  const float* __restrict__ xr = x + rowBase;
  float* __restrict__ outr = out + rowBase;

  // ---- Issue all async global->LDS copies up front (max MLP) ----
#if HAVE_ASYNC_LDS
  #pragma unroll
  for (int j = 0; j < VEC_PER_THREAD; ++j) {
    const int e = (tid + j * BLOCK) * 4;
    __builtin_amdgcn_global_load_async_to_lds_b128(
        (gv4i_p)(xr + e), (lv4i_p)(&tile[e]), 0, 0);
  }
#else
  #pragma unroll
  for (int j = 0; j < VEC_PER_THREAD; ++j) {
    const int e = (tid + j * BLOCK) * 4;
    *(v4f*)(&tile[e]) = *(const v4f*)(xr + e);
  }
#endif

  hist[tid] = 0u;
  hist[tid + BLOCK] = 0u;
  if (tid == 0) { s_above = 0u; s_claim = 0u; }
  __syncthreads();

  // ---- Pass 1: consume chunks as the DMA lands; keep keys in registers ----
  // Async loads complete in order per wave: after s_wait_asynccnt(7-J),
  // chunks 0..J of THIS wave (the only ones it reads in pass 1) are in LDS.
  unsigned kv[VEC_PER_THREAD][4];

#define PASS1_STEP(J)                                                         \
  {                                                                           \
    WAIT_ASYNC(VEC_PER_THREAD - 1 - (J));                                     \
    asm volatile("" ::: "memory"); /* keep the LDS read below the wait */     \
    const int e = (tid + (J) * BLOCK) * 4;                                    \
    const v4f v = *(const v4f*)(&tile[e]);                                    \
    kv[(J)][0] = f2key(v.x);                                                  \
    kv[(J)][1] = f2key(v.y);                                                  \
    kv[(J)][2] = f2key(v.z);                                                  \
    kv[(J)][3] = f2key(v.w);                                                  \
    atomicAdd(&hist[kv[(J)][0] >> 21], 1u);                                   \
    atomicAdd(&hist[kv[(J)][1] >> 21], 1u);                                   \
    atomicAdd(&hist[kv[(J)][2] >> 21], 1u);                                   \
    atomicAdd(&hist[kv[(J)][3] >> 21], 1u);                                   \
  }

  PASS1_STEP(0)
  PASS1_STEP(1)
  PASS1_STEP(2)
  PASS1_STEP(3)
  PASS1_STEP(4)
  PASS1_STEP(5)
  PASS1_STEP(6)
  PASS1_STEP(7)
#undef PASS1_STEP

  __syncthreads();
  select_bin(2048u, KSEL, hist, chunk, &s_bin, &s_above, tid);
  const unsigned b1 = s_bin;
  const unsigned want2 = KSEL - s_above;

  hist[tid] = 0u;
  hist[tid + BLOCK] = 0u;
  __syncthreads();

  // ---- Pass 2 (registers): among bin b1, histogram key[20:10] ----
  #pragma unroll
  for (int j = 0; j < VEC_PER_THREAD; ++j) {
    #pragma unroll
    for (int c = 0; c < 4; ++c) {
      const unsigned k = kv[j][c];
      if ((k >> 21) == b1) atomicAdd(&hist[(k >> 10) & 2047u], 1u);
    }
  }
  __syncthreads();
  select_bin(2048u, want2, hist, chunk, &s_bin, &s_above, tid);
  const unsigned pre22 = (b1 << 11) | s_bin;  // == key >> 10 of threshold
  const unsigned want3 = KSEL - s_above;

  hist[tid] = 0u;   // only low 1024 bins used next
  __syncthreads();

  // ---- Pass 3 (registers): among prefix pre22, histogram key[9:0] ----
  #pragma unroll
  for (int j = 0; j < VEC_PER_THREAD; ++j) {
    #pragma unroll
    for (int c = 0; c < 4; ++c) {
      const unsigned k = kv[j][c];
      if ((k >> 10) == pre22) atomicAdd(&hist[k & 1023u], 1u);
    }
  }
  __syncthreads();
  select_bin(1024u, want3, hist, chunk, &s_bin, &s_above, tid);
  const unsigned T = (pre22 << 10) | s_bin;     // threshold key (K-th largest)
  const unsigned tieBudget = KSEL - s_above;    // #elements == T to accept (>=1)

  // ---- Write pass (registers): keep top-K (ReLU'd), zero elsewhere ----
  #pragma unroll
  for (int j = 0; j < VEC_PER_THREAD; ++j) {
    const int e = (tid + j * BLOCK) * 4;
    v4f o;
    #pragma unroll
    for (int c = 0; c < 4; ++c) {
      const unsigned k = kv[j][c];
      float r = 0.0f;
      if (k > T) {
        r = fmaxf(key2f(k), 0.0f);
      } else if (k == T) {
        r = (atomicAdd(&s_claim, 1u) < tieBudget) ? fmaxf(key2f(k), 0.0f) : 0.0f;
      }
      o[c] = r;
    }
    __builtin_nontemporal_store(o, (v4f*)(outr + e));  // write-once stream
  }
}

extern "C" void kernel_launch(void* const* d_in, const int* in_sizes, int n_in,
                              void* d_out, int out_size, void* d_ws, size_t ws_size,
                              hipStream_t stream) {
  (void)n_in; (void)out_size; (void)d_ws; (void)ws_size;
  const float* x = (const float*)d_in[0];
  float* out = (float*)d_out;
  const int rows = in_sizes[0] / NCOLS;   // 4096 for the reference shapes
  topk_relu_scatter<<<dim3(rows), dim3(BLOCK), 0, stream>>>(x, out);
}